// FFN_GCNs_11871289606265
// MI455X (gfx1250) — compile-verified
//
#include <hip/hip_runtime.h>

typedef float v2f __attribute__((ext_vector_type(2)));
typedef float v8f __attribute__((ext_vector_type(8)));

#define N_NODES 50000
#define N_EDGES 800000
#define IN_DIM  64
#define HIDDEN  128
#define N_PAIRS 100000

// ---------------------------------------------------------------------------
// Degree / normalization:  deg includes self-loop (init to 1), dinv = rsqrt.
// ---------------------------------------------------------------------------
__global__ __launch_bounds__(256) void deg_init(float* __restrict__ deg) {
    int i = blockIdx.x * blockDim.x + threadIdx.x;
    if (i < N_NODES) deg[i] = 1.0f;           // self-loop contribution
}

__global__ __launch_bounds__(256) void deg_count(const int* __restrict__ ei,
                                                 float* __restrict__ deg) {
    int e = blockIdx.x * blockDim.x + threadIdx.x;
    if (e < N_EDGES) atomicAdd(&deg[ei[N_EDGES + e]], 1.0f);   // dst row
}

__global__ __launch_bounds__(256) void deg_to_dinv(float* __restrict__ dinv) {
    int i = blockIdx.x * blockDim.x + threadIdx.x;
    if (i < N_NODES) dinv[i] = rsqrtf(fmaxf(dinv[i], 1.0e-12f));
}

// ---------------------------------------------------------------------------
// Repack W[K x 128] into WMMA B-fragment order so the GEMM inner loop does a
// single contiguous b64 load per lane per WMMA step.
// Layout: Bp[(c * (K/4) + k4) * 32 + lane] = float2{ W[k+koff][c*16+n],
//                                                    W[k+koff+1][c*16+n] }
// with n = lane&15, koff = (lane>>4)*2, k = 4*k4, c = column tile (0..7).
// ---------------------------------------------------------------------------
__global__ __launch_bounds__(256)
void pack_W(const float* __restrict__ W, float* __restrict__ Bp, int K) {
    int idx   = blockIdx.x * blockDim.x + threadIdx.x;
    int steps = K >> 2;
    int total = 8 * steps * 32;
    if (idx >= total) return;
    int lane = idx & 31;
    int k4   = (idx >> 5) % steps;
    int c    = idx / (32 * steps);
    int n    = lane & 15;
    int koff = (lane >> 4) * 2;
    int k    = k4 * 4;
    v2f v;
    v.x = W[(size_t)(k + koff)     * HIDDEN + c * 16 + n];
    v.y = W[(size_t)(k + koff + 1) * HIDDEN + c * 16 + n];
    *(v2f*)(Bp + (size_t)idx * 2) = v;
}

// ---------------------------------------------------------------------------
// H[N x 128] = X[N x K] @ W[K x 128]   (fp32, V_WMMA_F32_16X16X4_F32)
// One wave per 16x16 output tile; 8 waves/block cover all 8 column tiles,
// blockIdx.x selects the row tile (N = 3125 * 16 exactly).
// A frag (16x4 f32): lane<16 -> K=k..k+1, lane>=16 -> K=k+2..k+3, M=lane&15.
// B frag comes prepacked (contiguous b64 per lane).
// C/D frag: vgpr i holds M = i + 8*(lane>=16), N = lane&15.
// ---------------------------------------------------------------------------
template <int K>
__global__ __launch_bounds__(256)
void gemm_xw_wmma(const float* __restrict__ X, const float* __restrict__ Bp,
                  float* __restrict__ H) {
    const int wave = threadIdx.x >> 5;       // 0..7 -> column tile
    const int lane = threadIdx.x & 31;
    const int row0 = blockIdx.x * 16;
    const int col0 = wave * 16;
    const int mn   = lane & 15;              // M for A-frag, N for D-frag
    const int koff = (lane >> 4) * 2;        // 0 or 2

    v8f acc = {0.f, 0.f, 0.f, 0.f, 0.f, 0.f, 0.f, 0.f};
    const float* xrow = X + (size_t)(row0 + mn) * K;
    const v2f*   bp   = (const v2f*)Bp + ((size_t)wave * (K / 4) * 32 + lane);

#pragma unroll 8
    for (int k4 = 0; k4 < K / 4; ++k4) {
        v2f a = *(const v2f*)(xrow + k4 * 4 + koff);   // contiguous K pair
        v2f b = bp[(size_t)k4 * 32];                   // prepacked B-frag
        acc = __builtin_amdgcn_wmma_f32_16x16x4_f32(
            /*neg_a=*/false, a, /*neg_b=*/false, b,
            /*c_mod=*/(short)0, acc, /*reuse_a=*/false, /*reuse_b=*/false);
    }

    const int mbase = (lane >> 4) * 8;
    float* hp = H + (size_t)row0 * HIDDEN + col0 + mn;
#pragma unroll
    for (int i = 0; i < 8; ++i)
        hp[(size_t)(mbase + i) * HIDDEN] = acc[i];
}

// ---------------------------------------------------------------------------
// Y[i,:] = dinv[i]^2 * H[i,:]   (self-loop term; also initializes Y)
// One thread per float4 (N*32 threads).
// ---------------------------------------------------------------------------
__global__ __launch_bounds__(256)
void agg_self(const float* __restrict__ H, const float* __restrict__ dinv,
              float* __restrict__ Y) {
    int idx = blockIdx.x * blockDim.x + threadIdx.x;
    if (idx >= N_NODES * 32) return;
    float d = dinv[idx >> 5];
    float w = d * d;
    float4 h = *(const float4*)(H + (size_t)idx * 4);
    float4 r = {h.x * w, h.y * w, h.z * w, h.w * w};
    *(float4*)(Y + (size_t)idx * 4) = r;
}

// ---------------------------------------------------------------------------
// Edge scatter: one wave per edge; 32 lanes x float4 = 128 features.
// Y[dst,:] += dinv[src]*dinv[dst] * H[src,:]  via global_atomic_add_f32.
// ---------------------------------------------------------------------------
__global__ __launch_bounds__(256)
void agg_edges(const float* __restrict__ H, const int* __restrict__ ei,
               const float* __restrict__ dinv, float* __restrict__ Y) {
    int wid  = (blockIdx.x * blockDim.x + threadIdx.x) >> 5;
    int lane = threadIdx.x & 31;
    if (wid >= N_EDGES) return;
    int src = ei[wid];
    int dst = ei[N_EDGES + wid];
    float w = dinv[src] * dinv[dst];
    float4 h = *(const float4*)(H + (size_t)src * HIDDEN + lane * 4);
    float* y = Y + (size_t)dst * HIDDEN + lane * 4;
    atomicAdd(y + 0, h.x * w);
    atomicAdd(y + 1, h.y * w);
    atomicAdd(y + 2, h.z * w);
    atomicAdd(y + 3, h.w * w);
}

// ---------------------------------------------------------------------------
// X = relu(Y + b)
// ---------------------------------------------------------------------------
__global__ __launch_bounds__(256)
void finalize_relu_bias(const float* __restrict__ Y, const float* __restrict__ b,
                        float* __restrict__ X) {
    int idx = blockIdx.x * blockDim.x + threadIdx.x;
    if (idx >= N_NODES * HIDDEN) return;
    float v = Y[idx] + b[idx & (HIDDEN - 1)];
    X[idx] = v > 0.f ? v : 0.f;
}

// ---------------------------------------------------------------------------
// out[p,:] = relu(concat(xl[a], xr[b]) @ fcW + fcb), one wave per pair.
// ---------------------------------------------------------------------------
__global__ __launch_bounds__(256)
void fc_pairs(const float* __restrict__ xl, const float* __restrict__ xr,
              const int* __restrict__ label, const float* __restrict__ fcW,
              const float* __restrict__ fcb, float* __restrict__ out) {
    int wid  = (blockIdx.x * blockDim.x + threadIdx.x) >> 5;
    int lane = threadIdx.x & 31;
    if (wid >= N_PAIRS) return;
    int ia = label[2 * wid];
    int ib = label[2 * wid + 1];
    const float* ra = xl + (size_t)ia * HIDDEN;
    const float* rb = xr + (size_t)ib * HIDDEN;
    float acc0 = 0.f, acc1 = 0.f;
#pragma unroll
    for (int t = 0; t < 4; ++t) {
        int f = lane + t * 32;             // 0..127
        float va = ra[f];
        float vb = rb[f];
        acc0 += va * fcW[f * 2 + 0] + vb * fcW[(f + 128) * 2 + 0];
        acc1 += va * fcW[f * 2 + 1] + vb * fcW[(f + 128) * 2 + 1];
    }
#pragma unroll
    for (int off = 16; off > 0; off >>= 1) {
        acc0 += __shfl_xor(acc0, off, 32);
        acc1 += __shfl_xor(acc1, off, 32);
    }
    if (lane == 0) {
        float o0 = acc0 + fcb[0];
        float o1 = acc1 + fcb[1];
        out[2 * wid + 0] = o0 > 0.f ? o0 : 0.f;
        out[2 * wid + 1] = o1 > 0.f ? o1 : 0.f;
    }
}

// ---------------------------------------------------------------------------
// Orchestration.  Workspace layout (floats):
//   [0]          xl   : N*128   (tower-1 activations, reused per layer)
//   [N*128]      xr   : N*128   (tower-2 activations)
//   [2*N*128]    H    : N*128   (GEMM output)
//   [3*N*128]    Y    : N*128   (aggregation accumulator)
//   [4*N*128]    dinv : N
//   [...]        Bp   : 16384   (prepacked W fragments, 64 KB)
// Total ~102.7 MB.
// ---------------------------------------------------------------------------
extern "C" void kernel_launch(void* const* d_in, const int* in_sizes, int n_in,
                              void* d_out, int out_size, void* d_ws, size_t ws_size,
                              hipStream_t stream) {
    const float* x1    = (const float*)d_in[0];
    const int*   ei1   = (const int*)  d_in[1];
    const float* x2    = (const float*)d_in[2];
    const int*   ei2   = (const int*)  d_in[3];
    const int*   label = (const int*)  d_in[4];
    const float* W0    = (const float*)d_in[5];
    const float* b0    = (const float*)d_in[6];
    const float* W1    = (const float*)d_in[7];
    const float* b1    = (const float*)d_in[8];
    const float* W2    = (const float*)d_in[9];
    const float* b2    = (const float*)d_in[10];
    const float* fcW   = (const float*)d_in[11];
    const float* fcb   = (const float*)d_in[12];
    float* out = (float*)d_out;

    float* ws   = (float*)d_ws;
    const size_t NF = (size_t)N_NODES * HIDDEN;
    float* xl   = ws;
    float* xr   = ws + NF;
    float* H    = ws + 2 * NF;
    float* Y    = ws + 3 * NF;
    float* dinv = ws + 4 * NF;
    float* Bp   = ws + 4 * NF + N_NODES;   // up to 8*(128/4)*32*2 = 16384 floats

    const int TPB = 256;
    const int gN     = (N_NODES + TPB - 1) / TPB;          // 196
    const int gE     = (N_EDGES + TPB - 1) / TPB;          // 3125
    const int gGemm  = N_NODES / 16;                       // 3125 row tiles
    const int gSelf  = (N_NODES * 32 + TPB - 1) / TPB;     // 6250
    const int gFin   = (N_NODES * HIDDEN + TPB - 1) / TPB; // 25000
    const int gEdgeW = (N_EDGES * 32 + TPB - 1) / TPB;     // 100000
    const int gPairW = (N_PAIRS * 32 + TPB - 1) / TPB;     // 12500

    const float* xin[2]  = {x1, x2};
    const int*   ein[2]  = {ei1, ei2};
    float*       xbuf[2] = {xl, xr};
    const float* Ws[3] = {W0, W1, W2};
    const float* bs[3] = {b0, b1, b2};

    for (int t = 0; t < 2; ++t) {
        // Symmetric normalization for this graph.
        deg_init   <<<gN, TPB, 0, stream>>>(dinv);
        deg_count  <<<gE, TPB, 0, stream>>>(ein[t], dinv);
        deg_to_dinv<<<gN, TPB, 0, stream>>>(dinv);

        for (int l = 0; l < 3; ++l) {
            const float* Xin = (l == 0) ? xin[t] : xbuf[t];
            const int    K   = (l == 0) ? IN_DIM : HIDDEN;
            const int    gPk = (8 * (K / 4) * 32 + TPB - 1) / TPB;
            pack_W<<<gPk, TPB, 0, stream>>>(Ws[l], Bp, K);
            if (K == IN_DIM)
                gemm_xw_wmma<IN_DIM><<<gGemm, TPB, 0, stream>>>(Xin, Bp, H);
            else
                gemm_xw_wmma<HIDDEN><<<gGemm, TPB, 0, stream>>>(Xin, Bp, H);
            agg_self <<<gSelf, TPB, 0, stream>>>(H, dinv, Y);
            agg_edges<<<gEdgeW, TPB, 0, stream>>>(H, ein[t], dinv, Y);
            finalize_relu_bias<<<gFin, TPB, 0, stream>>>(Y, bs[l], xbuf[t]);
        }
    }

    fc_pairs<<<gPairW, TPB, 0, stream>>>(xl, xr, label, fcW, fcb, out);
}